// SphericalGraphConvBlock_4071628997386
// MI455X (gfx1250) — compile-verified
//
#include <hip/hip_runtime.h>
#include <hip/hip_bf16.h>

typedef float v2f __attribute__((ext_vector_type(2)));
typedef float v8f __attribute__((ext_vector_type(8)));

#define NB   8
#define CIN  64
#define COUT 128
#define KTAPS 5
#define NN   49152
#define NE   393216
#define BN_EPS 1e-5f

// ---------------- graph normalization ----------------
__global__ void deg_kernel(const int* __restrict__ ei, const float* __restrict__ ew,
                           float* __restrict__ deg) {
  int e = blockIdx.x * blockDim.x + threadIdx.x;
  if (e >= NE) return;
  int s = ei[e], d = ei[NE + e];
  float w = (s == d) ? 0.f : ew[e];
  if (w != 0.f) atomicAdd(&deg[s], w);
}

__global__ void dinv_kernel(float* deg) {
  int i = blockIdx.x * blockDim.x + threadIdx.x;
  if (i >= NN) return;
  float d = deg[i];
  deg[i] = (d > 0.f) ? rsqrtf(d) : 0.f;
}

__global__ void norm_count_kernel(const int* __restrict__ ei, const float* __restrict__ ew,
                                  const float* __restrict__ dinv, float* __restrict__ norm,
                                  int* __restrict__ cnt) {
  int e = blockIdx.x * blockDim.x + threadIdx.x;
  if (e >= NE) return;
  int s = ei[e], d = ei[NE + e];
  float w = (s == d) ? 0.f : ew[e];
  norm[e] = -dinv[s] * w * dinv[d];
  atomicAdd(&cnt[d], 1);
}

// single-block exclusive scan over NN counts -> row_ptr[NN+1]
__global__ void scan_kernel(const int* __restrict__ cnt, int* __restrict__ row_ptr) {
  __shared__ int sh[1024];
  __shared__ int carry;
  if (threadIdx.x == 0) carry = 0;
  __syncthreads();
  for (int base = 0; base < NN; base += 1024) {
    int i = base + threadIdx.x;
    int v = (i < NN) ? cnt[i] : 0;
    sh[threadIdx.x] = v;
    __syncthreads();
    for (int off = 1; off < 1024; off <<= 1) {
      int t = (threadIdx.x >= off) ? sh[threadIdx.x - off] : 0;
      __syncthreads();
      sh[threadIdx.x] += t;
      __syncthreads();
    }
    if (i < NN) row_ptr[i] = carry + sh[threadIdx.x] - v;
    __syncthreads();
    if (threadIdx.x == 0) carry += sh[1023];
    __syncthreads();
  }
  if (threadIdx.x == 0) row_ptr[NN] = carry;
}

__global__ void fill_csr_kernel(const int* __restrict__ ei, const float* __restrict__ norm,
                                const int* __restrict__ row_ptr, int* __restrict__ fill,
                                int* __restrict__ csr_src, float* __restrict__ csr_val) {
  int e = blockIdx.x * blockDim.x + threadIdx.x;
  if (e >= NE) return;
  int s = ei[e], d = ei[NE + e];
  int pos = row_ptr[d] + atomicAdd(&fill[d], 1);
  csr_src[pos] = s;
  csr_val[pos] = norm[e];
}

// ---------------- x (B,Cin,N) -> T0 (B,N,Cin), LDS-tiled ----------------
__global__ void transpose_x_kernel(const float* __restrict__ x, float* __restrict__ T0) {
  __shared__ float tile[64][33];
  int b = blockIdx.y;
  int n0 = blockIdx.x * 32;
  int tn = threadIdx.x & 31;
  int c0 = threadIdx.x >> 5;        // 0..7
  for (int cc = c0; cc < 64; cc += 8)
    tile[cc][tn] = x[((size_t)b * CIN + cc) * NN + n0 + tn];
  __syncthreads();
  int c  = threadIdx.x & 63;
  int nl = threadIdx.x >> 6;        // 0..3
  for (int nn = nl; nn < 32; nn += 4)
    T0[((size_t)b * NN + n0 + nn) * CIN + c] = tile[c][nn];
}

// ---------------- CSR gather propagation; Tout = alpha*agg + beta*Tprev ----------------
__global__ void prop_kernel(const float* __restrict__ Tin, const float* __restrict__ Tprev,
                            float* __restrict__ Tout,
                            const int* __restrict__ row_ptr, const int* __restrict__ csr_src,
                            const float* __restrict__ csr_val, float alpha, float beta) {
  int wave = (blockIdx.x * blockDim.x + threadIdx.x) >> 5;
  int lane = threadIdx.x & 31;
  if (wave >= NN * NB) return;
  int node = wave % NN;
  int b    = wave / NN;
  const float* Tb = Tin + (size_t)b * NN * CIN;
  int e0 = row_ptr[node], e1 = row_ptr[node + 1];
  float a0 = 0.f, a1 = 0.f;
  for (int j = e0; j < e1; ++j) {
    int s = csr_src[j];
    float v = csr_val[j];
    if (j + 1 < e1) __builtin_prefetch(Tb + (size_t)csr_src[j + 1] * CIN, 0, 3);
    const float* row = Tb + (size_t)s * CIN;
    a0 += v * row[lane];
    a1 += v * row[lane + 32];
  }
  size_t o = ((size_t)b * NN + node) * CIN;
  float r0 = alpha * a0, r1 = alpha * a1;
  if (beta != 0.f) { r0 += beta * Tprev[o + lane]; r1 += beta * Tprev[o + lane + 32]; }
  Tout[o + lane]      = r0;
  Tout[o + lane + 32] = r1;
}

// ---------------- O[r,0:128] (+)= T[r,0:64] @ W(64x128) via V_WMMA_F32_16X16X4_F32 ----------------
__global__ void __launch_bounds__(256) cheb_gemm_kernel(const float* __restrict__ T,
                                                        const float* __restrict__ W,
                                                        float* __restrict__ O,
                                                        const float* __restrict__ bias,
                                                        int accumulate) {
  __shared__ float sW[CIN * COUT];                     // 32 KB
  for (int i = threadIdx.x; i < CIN * COUT; i += 256) sW[i] = W[i];
  __syncthreads();

  int w    = threadIdx.x >> 5;                         // wave 0..7 -> 16-row slice
  int lane = threadIdx.x & 31;
  int half = lane >> 4;                                // 0/1
  int m    = lane & 15;
  int row  = blockIdx.x * 128 + w * 16 + m;            // this lane's A row
  const float* Arow = T + (size_t)row * CIN + 2 * half;
  size_t obase = ((size_t)blockIdx.x * 128 + w * 16) * COUT;

  v8f acc[8];
  if (accumulate) {
    for (int t = 0; t < 8; ++t)
      for (int v = 0; v < 8; ++v)
        acc[t][v] = O[obase + (size_t)(v + 8 * half) * COUT + 16 * t + m];
  } else {
    for (int t = 0; t < 8; ++t)
      for (int v = 0; v < 8; ++v) acc[t][v] = 0.f;
  }

  for (int kk = 0; kk < CIN; kk += 4) {
    v2f a;                                             // A 16x4: lane<16 K={kk,kk+1}, lane>=16 K={kk+2,kk+3}
    a.x = Arow[kk];
    a.y = Arow[kk + 1];
    int brow = kk + 2 * half;                          // B 4x16: VGPR0 K=kk(+2), VGPR1 K=kk+1(+3)
    for (int t = 0; t < 8; ++t) {
      v2f bb;
      bb.x = sW[brow * COUT + 16 * t + m];
      bb.y = sW[(brow + 1) * COUT + 16 * t + m];
      acc[t] = __builtin_amdgcn_wmma_f32_16x16x4_f32(false, a, false, bb,
                                                     (short)0, acc[t], false, false);
    }
  }

  for (int t = 0; t < 8; ++t) {
    float badd = bias ? bias[16 * t + m] : 0.f;
    for (int v = 0; v < 8; ++v)
      O[obase + (size_t)(v + 8 * half) * COUT + 16 * t + m] = acc[t][v] + badd;
  }
}

// ---------------- BN stats: column j = group mod 128; one wave per 128-float group ----------------
__global__ void bn_stats_kernel(const float* __restrict__ O, float* __restrict__ gsum,
                                float* __restrict__ gsq) {
  __shared__ float ls[COUT], lq[COUT];
  if (threadIdx.x < COUT) { ls[threadIdx.x] = 0.f; lq[threadIdx.x] = 0.f; }
  __syncthreads();
  int lane   = threadIdx.x & 31;
  int wave   = (blockIdx.x * blockDim.x + threadIdx.x) >> 5;
  int nwaves = (gridDim.x * blockDim.x) >> 5;
  const int groups = NB * NN;
  for (int g = wave; g < groups; g += nwaves) {
    float4 v = ((const float4*)(O + (size_t)g * COUT))[lane];
    float s = v.x + v.y + v.z + v.w;
    float q = v.x * v.x + v.y * v.y + v.z * v.z + v.w * v.w;
    for (int off = 16; off > 0; off >>= 1) {
      s += __shfl_down(s, off, 32);
      q += __shfl_down(q, off, 32);
    }
    if (lane == 0) {
      int j = g & (COUT - 1);
      atomicAdd(&ls[j], s);
      atomicAdd(&lq[j], q);
    }
  }
  __syncthreads();
  if (threadIdx.x < COUT) {
    atomicAdd(&gsum[threadIdx.x], ls[threadIdx.x]);
    atomicAdd(&gsq[threadIdx.x],  lq[threadIdx.x]);
  }
}

__global__ void bn_finalize_kernel(const float* __restrict__ gsum, const float* __restrict__ gsq,
                                   const float* __restrict__ gamma, const float* __restrict__ beta,
                                   float* __restrict__ scale, float* __restrict__ shift) {
  int j = threadIdx.x;
  const float invc = 1.f / (float)(NB * NN);
  float mean = gsum[j] * invc;
  float var  = gsq[j] * invc - mean * mean;
  float inv  = rsqrtf(var + BN_EPS);
  float sc   = gamma[j] * inv;
  scale[j] = sc;
  shift[j] = beta[j] - mean * sc;
}

// ---------------- BN apply + output permutation + ReLU ----------------
// d_out[b][c][n] = relu(scale[c]*O[b][n'][c'] + shift[c]), r=n*128+c, c'=r/N, n'=r%N
__global__ void bn_permute_kernel(const float* __restrict__ O, const float* __restrict__ scale,
                                  const float* __restrict__ shift, float* __restrict__ out) {
  __shared__ float tile[32 * 385];                     // ~48 KB, pad avoids bank conflicts
  int b = blockIdx.z, c = blockIdx.y, cp0 = blockIdx.x * 32;
  const float* Ob = O + (size_t)b * NN * COUT;
  int lane = threadIdx.x & 31;
  int w    = threadIdx.x >> 5;
  // source element (n' = c + 128*m, c' = cp0+lane) -> tile[lane][m]
  for (int m = w; m < 384; m += 8)
    tile[lane * 385 + m] = Ob[(size_t)(c + 128 * m) * COUT + cp0 + lane];
  __syncthreads();
  float sc = scale[c], sh = shift[c];
  float* dst = out + ((size_t)b * COUT + c) * NN + (size_t)cp0 * 384;  // contiguous 12288 floats
  for (int i = threadIdx.x; i < 32 * 384; i += 256) {
    int cp = i / 384, m = i - cp * 384;                // dest n = (cp0+cp)*384 + m
    float v = tile[cp * 385 + m] * sc + sh;
    dst[i] = v > 0.f ? v : 0.f;
  }
}

extern "C" void kernel_launch(void* const* d_in, const int* in_sizes, int n_in,
                              void* d_out, int out_size, void* d_ws, size_t ws_size,
                              hipStream_t stream) {
  (void)in_sizes; (void)n_in; (void)out_size; (void)ws_size;
  const float* x     = (const float*)d_in[0];
  const int*   ei    = (const int*)  d_in[1];
  const float* ew    = (const float*)d_in[2];
  const float* cw    = (const float*)d_in[3];
  const float* bias  = (const float*)d_in[4];
  const float* gamma = (const float*)d_in[5];
  const float* beta  = (const float*)d_in[6];
  float* out = (float*)d_out;

  char* ws = (char*)d_ws;
  size_t off = 0;
  auto carve = [&](size_t bytes) -> void* {
    void* p = ws + off;
    off = (off + bytes + 255) & ~(size_t)255;
    return p;
  };
  float* deg     = (float*)carve((size_t)NN * 4);
  float* norm    = (float*)carve((size_t)NE * 4);
  int*   row_ptr = (int*)  carve((size_t)(NN + 1) * 4);
  int*   fill    = (int*)  carve((size_t)NN * 4);
  int*   csr_src = (int*)  carve((size_t)NE * 4);
  float* csr_val = (float*)carve((size_t)NE * 4);
  float* gsum    = (float*)carve((size_t)COUT * 4);
  float* gsq     = (float*)carve((size_t)COUT * 4);
  float* scale   = (float*)carve((size_t)COUT * 4);
  float* shift   = (float*)carve((size_t)COUT * 4);
  const size_t tbytes = (size_t)NB * NN * CIN * 4;     // 100.7 MB each
  float* T0 = (float*)carve(tbytes);
  float* T1 = (float*)carve(tbytes);
  float* T2 = (float*)carve(tbytes);
  float* O  = (float*)carve((size_t)NB * NN * COUT * 4);

  const int EB = (NE + 255) / 256;
  hipMemsetAsync(deg,  0, (size_t)NN * 4, stream);
  hipMemsetAsync(fill, 0, (size_t)NN * 4, stream);
  hipMemsetAsync(gsum, 0, (size_t)COUT * 4, stream);
  hipMemsetAsync(gsq,  0, (size_t)COUT * 4, stream);

  deg_kernel<<<EB, 256, 0, stream>>>(ei, ew, deg);
  dinv_kernel<<<(NN + 255) / 256, 256, 0, stream>>>(deg);
  norm_count_kernel<<<EB, 256, 0, stream>>>(ei, ew, deg, norm, fill);
  scan_kernel<<<1, 1024, 0, stream>>>(fill, row_ptr);
  hipMemsetAsync(fill, 0, (size_t)NN * 4, stream);
  fill_csr_kernel<<<EB, 256, 0, stream>>>(ei, norm, row_ptr, fill, csr_src, csr_val);

  transpose_x_kernel<<<dim3(NN / 32, NB), 256, 0, stream>>>(x, T0);

  const int GEMM_BLOCKS = (NB * NN) / 128;             // 3072
  const int PROP_BLOCKS = (NB * NN) / 8;               // 49152 (8 waves/block)
  const size_t WK = (size_t)CIN * COUT;                // 8192 floats per tap

  // k=0: out = Tx0 @ W0
  cheb_gemm_kernel<<<GEMM_BLOCKS, 256, 0, stream>>>(T0, cw, O, nullptr, 0);
  // Tx1 = prop(Tx0); out += Tx1 @ W1
  prop_kernel<<<PROP_BLOCKS, 256, 0, stream>>>(T0, T0, T1, row_ptr, csr_src, csr_val, 1.f, 0.f);
  cheb_gemm_kernel<<<GEMM_BLOCKS, 256, 0, stream>>>(T1, cw + 1 * WK, O, nullptr, 1);
  // Tx2 = 2*prop(Tx1) - Tx0
  prop_kernel<<<PROP_BLOCKS, 256, 0, stream>>>(T1, T0, T2, row_ptr, csr_src, csr_val, 2.f, -1.f);
  cheb_gemm_kernel<<<GEMM_BLOCKS, 256, 0, stream>>>(T2, cw + 2 * WK, O, nullptr, 1);
  // Tx3 = 2*prop(Tx2) - Tx1  (reuse T0 buffer)
  prop_kernel<<<PROP_BLOCKS, 256, 0, stream>>>(T2, T1, T0, row_ptr, csr_src, csr_val, 2.f, -1.f);
  cheb_gemm_kernel<<<GEMM_BLOCKS, 256, 0, stream>>>(T0, cw + 3 * WK, O, nullptr, 1);
  // Tx4 = 2*prop(Tx3) - Tx2  (reuse T1 buffer); fuse bias into last pass
  prop_kernel<<<PROP_BLOCKS, 256, 0, stream>>>(T0, T2, T1, row_ptr, csr_src, csr_val, 2.f, -1.f);
  cheb_gemm_kernel<<<GEMM_BLOCKS, 256, 0, stream>>>(T1, cw + 4 * WK, O, bias, 1);

  bn_stats_kernel<<<1024, 256, 0, stream>>>(O, gsum, gsq);
  bn_finalize_kernel<<<1, COUT, 0, stream>>>(gsum, gsq, gamma, beta, scale, shift);
  bn_permute_kernel<<<dim3(4, COUT, NB), 256, 0, stream>>>(O, scale, shift, out);
}